// ODEBlock_82179904241691
// MI455X (gfx1250) — compile-verified
//
#include <hip/hip_runtime.h>
#include <math.h>

// ---------------------------------------------------------------------------
// Problem constants (from reference): y0 [64,64,32,32] f32, w [64,64,3,3].
// ---------------------------------------------------------------------------
#define BB 64
#define CC 64
#define HH 32
#define WW 32
#define NTOT (BB * CC * HH * WW)        // 4,194,304 elements
#define N4   (NTOT / 4)
#define WPK_ELEMS (9 * 16 * 32 * 8)     // packed weight fragments per conv

// DoPri5 error coefficients (b5 - b4)
#define E1f (float)(35.0/384.0 - 5179.0/57600.0)
#define E3f (float)(500.0/1113.0 - 7571.0/16695.0)
#define E4f (float)(125.0/192.0 - 393.0/640.0)
#define E5f (float)(-2187.0/6784.0 + 92097.0/339200.0)
#define E6f (float)(11.0/84.0 - 187.0/2100.0)
#define E7f (float)(-1.0/40.0)

typedef float v2f __attribute__((ext_vector_type(2)));
typedef float v8f __attribute__((ext_vector_type(8)));

// ---------------------------------------------------------------------------
// Weight pre-pack: wpk[(((khkw*16 + c4)*32 + lane)*8) + mt*2 + v]
//   = wgt[co = mt*16 + (lane&15)][ci = c4*4 + 2*(lane>>4) + v][kh][kw]
// so each lane's 4 A-fragments for one K-chunk are 8 contiguous floats
// (two global_load_b128 in the conv inner loop).
// ---------------------------------------------------------------------------
__global__ __launch_bounds__(256) void pack_weights(
    const float* __restrict__ wgt, float* __restrict__ wpk)
{
    const int i = blockIdx.x * blockDim.x + threadIdx.x;
    if (i >= WPK_ELEMS) return;
    const int v    = i & 1;
    const int mt   = (i >> 1) & 3;
    const int lane = (i >> 3) & 31;
    const int c4   = (i >> 8) & 15;
    const int khkw = i >> 12;
    const int ln16 = lane & 15, half = lane >> 4;
    const int co = mt * 16 + ln16;
    const int ci = c4 * 4 + 2 * half + v;
    const int kh = khkw / 3, kw = khkw % 3;
    wpk[i] = wgt[((co * CC + ci) * 3 + kh) * 3 + kw];
}

// ---------------------------------------------------------------------------
// Implicit-GEMM 3x3 SAME conv using V_WMMA_F32_16X16X4_F32.
// Block = 8 waves = 8 consecutive output rows (same b, same 16-col tile).
// Input halo tile staged in LDS as [row 0..9][col 0..17][ci 0..63] (col
// stride 66 floats for bank spread), channels contiguous so a B fragment is
// one ds_load_b64. Staging uses GLOBAL_LOAD_ASYNC_TO_LDS_B32 (ASYNCcnt);
// out-of-image halo elements are zeroed with plain DS stores.
// VGPR striping per CDNA5 ISA 7.12.2:
//   A 16x4 : M = lane&15, a[v] holds K = v + 2*(lane>>4)
//   B 4x16 : N = lane&15, b[v] holds K = v + 2*(lane>>4)
//   C 16x16: c[r]: M = r + 8*(lane>>4), N = lane&15
// ---------------------------------------------------------------------------
#define LDS_COLSTRIDE 66
#define LDS_FLOATS (10 * 18 * LDS_COLSTRIDE)   // 11880 floats = 47520 B

template <bool RELU>
__global__ __launch_bounds__(256) void conv3x3_wmma(
    const float* __restrict__ x,     // [B,C,H,W]
    const float* __restrict__ wpk,   // packed weights (see pack_weights)
    const float* __restrict__ bias,  // [Cout]
    float* __restrict__ out)         // [B,Cout,H,W]
{
    __shared__ float lds[LDS_FLOATS];

    const int tid  = threadIdx.x;
    const int wid  = tid >> 5;                   // wave id = row within block
    const int lane = tid & 31;
    const int hb = blockIdx.x & 3;               // 4 row-blocks of 8
    const int wt = (blockIdx.x >> 2) & 1;        // 2 col tiles of 16
    const int b  = blockIdx.x >> 3;              // batch
    const int h0 = hb * 8, w0 = wt * 16;

    const float* __restrict__ xb = x + (size_t)b * (CC * HH * WW);

    // ---- stage [10][18][64] halo tile into LDS (async), zero the halo ----
    for (int f = tid; f < 10 * 18 * 64; f += 256) {
        const int col = f % 18;
        const int t2  = f / 18;
        const int ci  = t2 & 63;
        const int row = t2 >> 6;
        const int ih = h0 - 1 + row, iw = w0 - 1 + col;
        const int lidx = (row * 18 + col) * LDS_COLSTRIDE + ci;
        if ((unsigned)ih < (unsigned)HH && (unsigned)iw < (unsigned)WW) {
            // Generic address of __shared__ carries the LDS byte offset in
            // its low 32 bits (ISA 10.2 aperture rule).
            const unsigned loff = (unsigned)(uintptr_t)&lds[lidx];
            const float* gp = &xb[(ci * HH + ih) * WW + iw];
            asm volatile("global_load_async_to_lds_b32 %0, %1, off"
                         :: "v"(loff), "v"(gp) : "memory");
        } else {
            lds[lidx] = 0.0f;
        }
    }
    asm volatile("s_wait_asynccnt 0x0" ::: "memory");
    __syncthreads();

    const int ln16 = lane & 15;
    const int half = lane >> 4;

    v8f acc[4];
#pragma unroll
    for (int mt = 0; mt < 4; ++mt)
#pragma unroll
        for (int r = 0; r < 8; ++r) acc[mt][r] = 0.0f;

    const float4* __restrict__ wp4 = (const float4*)wpk;

#pragma unroll
    for (int kh = 0; kh < 3; ++kh) {
        const int row = wid + kh;                 // LDS row for ih = h+kh-1
#pragma unroll
        for (int kw = 0; kw < 3; ++kw) {
            const int khkw = kh * 3 + kw;
            const int col  = ln16 + kw;           // LDS col for iw = w+kw-1
            const float* bbase =
                &lds[(row * 18 + col) * LDS_COLSTRIDE + 2 * half];
            // float4 index of this lane's fragment block for c4 = 0
            const float4* abase = wp4 + ((size_t)(khkw * 16) * 32 + lane) * 2;

            for (int c4 = 0; c4 < 16; ++c4) {
                const float4 a01 = abase[(size_t)c4 * 64 + 0]; // mt0, mt1
                const float4 a23 = abase[(size_t)c4 * 64 + 1]; // mt2, mt3
                const v2f bf = *(const v2f*)(bbase + c4 * 4);  // ds_load_b64
                v2f af;
                af.x = a01.x; af.y = a01.y;
                acc[0] = __builtin_amdgcn_wmma_f32_16x16x4_f32(
                    false, af, false, bf, (short)0, acc[0], false, false);
                af.x = a01.z; af.y = a01.w;
                acc[1] = __builtin_amdgcn_wmma_f32_16x16x4_f32(
                    false, af, false, bf, (short)0, acc[1], false, false);
                af.x = a23.x; af.y = a23.y;
                acc[2] = __builtin_amdgcn_wmma_f32_16x16x4_f32(
                    false, af, false, bf, (short)0, acc[2], false, false);
                af.x = a23.z; af.y = a23.w;
                acc[3] = __builtin_amdgcn_wmma_f32_16x16x4_f32(
                    false, af, false, bf, (short)0, acc[3], false, false);
            }
        }
    }

    // ---- epilogue: bias (+ReLU), scatter to NCHW ----
    const int h = h0 + wid;
    const int w = w0 + ln16;
#pragma unroll
    for (int mt = 0; mt < 4; ++mt) {
#pragma unroll
        for (int r = 0; r < 8; ++r) {
            const int co = mt * 16 + r + 8 * half;
            float v = acc[mt][r] + bias[co];
            if (RELU) v = fmaxf(v, 0.0f);
            out[(((size_t)b * CC + co) * HH + h) * WW + w] = v;
        }
    }
}

// ---------------------------------------------------------------------------
// state[] layout (device): 0=t, 1=h, 2=hs (current clipped step), 3=accept&&!done
// ---------------------------------------------------------------------------
__global__ void init_state(float* state, float h0)
{
    state[0] = 0.0f; state[1] = h0; state[2] = 0.0f; state[3] = 0.0f;
}

__global__ void stage_begin(float* state)
{
    const float t = state[0], h = state[1];
    state[2] = fmaxf(0.0f, fminf(h, 1.0f - t));   // hs = min(h, t1 - t)
}

// out = y + hs * (c1*p1 + ... + c5*p5); unused slots pass nullptr
__global__ __launch_bounds__(256) void combine_kernel(
    const float4* __restrict__ y, float4* __restrict__ out,
    const float4* __restrict__ p1, float c1,
    const float4* __restrict__ p2, float c2,
    const float4* __restrict__ p3, float c3,
    const float4* __restrict__ p4, float c4,
    const float4* __restrict__ p5, float c5,
    const float* __restrict__ state, int n4)
{
    const float hs = state[2];
    int i = blockIdx.x * blockDim.x + threadIdx.x;
    const int stride = gridDim.x * blockDim.x;
    for (; i < n4; i += stride) {
        float4 r = y[i];
        float ax = 0.f, ay = 0.f, az = 0.f, aw = 0.f;
        if (p1) { float4 v = p1[i]; ax += c1 * v.x; ay += c1 * v.y; az += c1 * v.z; aw += c1 * v.w; }
        if (p2) { float4 v = p2[i]; ax += c2 * v.x; ay += c2 * v.y; az += c2 * v.z; aw += c2 * v.w; }
        if (p3) { float4 v = p3[i]; ax += c3 * v.x; ay += c3 * v.y; az += c3 * v.z; aw += c3 * v.w; }
        if (p4) { float4 v = p4[i]; ax += c4 * v.x; ay += c4 * v.y; az += c4 * v.z; aw += c4 * v.w; }
        if (p5) { float4 v = p5[i]; ax += c5 * v.x; ay += c5 * v.y; az += c5 * v.z; aw += c5 * v.w; }
        r.x += hs * ax; r.y += hs * ay; r.z += hs * az; r.w += hs * aw;
        out[i] = r;
    }
}

// Pass 1 of deterministic error-norm reduction: one partial per block.
__global__ __launch_bounds__(256) void err_partial_kernel(
    const float* __restrict__ y,  const float* __restrict__ y5,
    const float* __restrict__ k1, const float* __restrict__ k3,
    const float* __restrict__ k4, const float* __restrict__ k5,
    const float* __restrict__ k6, const float* __restrict__ k7,
    const float* __restrict__ state, float* __restrict__ partials, int n)
{
    const float hs = state[2];
    float s = 0.0f;
    int i = blockIdx.x * blockDim.x + threadIdx.x;
    const int stride = gridDim.x * blockDim.x;
    for (; i < n; i += stride) {
        const float e  = hs * (E1f * k1[i] + E3f * k3[i] + E4f * k4[i] +
                               E5f * k5[i] + E6f * k6[i] + E7f * k7[i]);
        const float sc = 1e-6f + 1e-5f * fmaxf(fabsf(y[i]), fabsf(y5[i]));
        const float r  = e / sc;
        s += r * r;
    }
    __shared__ float red[256];
    red[threadIdx.x] = s;
    __syncthreads();
    for (int o = 128; o > 0; o >>= 1) {
        if (threadIdx.x < o) red[threadIdx.x] += red[threadIdx.x + o];
        __syncthreads();
    }
    if (threadIdx.x == 0) partials[blockIdx.x] = red[0];
}

// Pass 2 + controller: finish reduction deterministically, update t/h/accept.
__global__ __launch_bounds__(256) void control_kernel(
    const float* __restrict__ partials, float* state, int nparts, float invN)
{
    __shared__ float red[256];
    float s = 0.0f;
    for (int i = threadIdx.x; i < nparts; i += 256) s += partials[i];
    red[threadIdx.x] = s;
    __syncthreads();
    for (int o = 128; o > 0; o >>= 1) {
        if (threadIdx.x < o) red[threadIdx.x] += red[threadIdx.x + o];
        __syncthreads();
    }
    if (threadIdx.x == 0) {
        const float t = state[0], h = state[1], hs = state[2];
        const float err_norm = sqrtf(red[0] * invN);
        const bool done   = (t >= 1.0f);
        const bool accept = (err_norm <= 1.0f);
        float fac = 0.9f * powf(fmaxf(err_norm, 1e-10f), -0.2f);
        fac = fminf(fmaxf(fac, 0.2f), 10.0f);
        state[0] = done ? t : (accept ? t + hs : t);
        state[1] = done ? h : hs * fac;
        state[3] = (!done && accept) ? 1.0f : 0.0f;
    }
}

// On accept: y <- y5, k1 <- k7 (FSAL). Uniform device-side predicate.
__global__ __launch_bounds__(256) void accept_kernel(
    float4* __restrict__ y, const float4* __restrict__ y5,
    float4* __restrict__ k1, const float4* __restrict__ k7,
    const float* __restrict__ state, int n4)
{
    if (state[3] == 0.0f) return;
    int i = blockIdx.x * blockDim.x + threadIdx.x;
    const int stride = gridDim.x * blockDim.x;
    for (; i < n4; i += stride) {
        y[i]  = y5[i];
        k1[i] = k7[i];
    }
}

// ---------------------------------------------------------------------------
// Host-side driver: fixed 48-step launch sequence, graph-capture safe.
// ---------------------------------------------------------------------------
extern "C" void kernel_launch(void* const* d_in, const int* in_sizes, int n_in,
                              void* d_out, int out_size, void* d_ws, size_t ws_size,
                              hipStream_t stream)
{
    (void)in_sizes; (void)n_in; (void)out_size; (void)ws_size;

    const float* y0 = (const float*)d_in[0];
    const float* w1 = (const float*)d_in[1];
    const float* b1 = (const float*)d_in[2];
    const float* w2 = (const float*)d_in[3];
    const float* b2 = (const float*)d_in[4];

    const size_t N = (size_t)NTOT;
    float* ws   = (float*)d_ws;
    float* y    = ws;
    float* k1   = ws + 1 * N;
    float* k2   = ws + 2 * N;
    float* k3   = ws + 3 * N;
    float* k4   = ws + 4 * N;
    float* k5   = ws + 5 * N;
    float* k6   = ws + 6 * N;
    float* k7   = ws + 7 * N;
    float* ytmp = ws + 8 * N;
    float* ztmp = ws + 9 * N;
    float* partials = ws + 10 * N;          // 1024 used
    float* state    = partials + 2048;
    float* wpk1     = state + 16;
    float* wpk2     = wpk1 + WPK_ELEMS;

    const dim3 cgrid(512), cblk(256);        // 512 blocks x 8 row-waves
    const dim3 egrid(2048), eblk(256);       // elementwise
    const int  RPARTS = 1024;                // reduction pass-1 blocks

    auto rhs = [&](const float* yin, float* kout) {
        conv3x3_wmma<true ><<<cgrid, cblk, 0, stream>>>(yin, wpk1, b1, ztmp);
        conv3x3_wmma<false><<<cgrid, cblk, 0, stream>>>(ztmp, wpk2, b2, kout);
    };
    auto comb = [&](const float* yin, float* out,
                    const float* p1, float c1, const float* p2, float c2,
                    const float* p3, float c3, const float* p4, float c4,
                    const float* p5, float c5) {
        combine_kernel<<<egrid, eblk, 0, stream>>>(
            (const float4*)yin, (float4*)out,
            (const float4*)p1, c1, (const float4*)p2, c2, (const float4*)p3, c3,
            (const float4*)p4, c4, (const float4*)p5, c5, state, N4);
    };

    // init: pack weights; y = y0; k1 = f(0, y0); state = {t=0, h=H0}
    pack_weights<<<dim3(WPK_ELEMS / 256), dim3(256), 0, stream>>>(w1, wpk1);
    pack_weights<<<dim3(WPK_ELEMS / 256), dim3(256), 0, stream>>>(w2, wpk2);
    hipMemcpyAsync(y, y0, N * sizeof(float), hipMemcpyDeviceToDevice, stream);
    init_state<<<1, 1, 0, stream>>>(state, 0.05f);
    rhs(y, k1);

    for (int step = 0; step < 48; ++step) {
        stage_begin<<<1, 1, 0, stream>>>(state);

        float* y5 = (float*)d_out;           // stage-5 solution scratch

        comb(y, ytmp, k1, 1.0f/5.0f, nullptr,0, nullptr,0, nullptr,0, nullptr,0);
        rhs(ytmp, k2);

        comb(y, ytmp, k1, 3.0f/40.0f, k2, 9.0f/40.0f, nullptr,0, nullptr,0, nullptr,0);
        rhs(ytmp, k3);

        comb(y, ytmp, k1, 44.0f/45.0f, k2, -56.0f/15.0f, k3, 32.0f/9.0f,
             nullptr,0, nullptr,0);
        rhs(ytmp, k4);

        comb(y, ytmp, k1, (float)(19372.0/6561.0), k2, (float)(-25360.0/2187.0),
             k3, (float)(64448.0/6561.0), k4, (float)(-212.0/729.0), nullptr,0);
        rhs(ytmp, k5);

        comb(y, ytmp, k1, (float)(9017.0/3168.0), k2, (float)(-355.0/33.0),
             k3, (float)(46732.0/5247.0), k4, (float)(49.0/176.0),
             k5, (float)(-5103.0/18656.0));
        rhs(ytmp, k6);

        comb(y, y5, k1, (float)(35.0/384.0), k3, (float)(500.0/1113.0),
             k4, (float)(125.0/192.0), k5, (float)(-2187.0/6784.0),
             k6, (float)(11.0/84.0));
        rhs(y5, k7);   // FSAL

        err_partial_kernel<<<dim3(RPARTS), dim3(256), 0, stream>>>(
            y, y5, k1, k3, k4, k5, k6, k7, state, partials, (int)NTOT);
        control_kernel<<<1, 256, 0, stream>>>(partials, state, RPARTS,
                                              1.0f / (float)NTOT);
        accept_kernel<<<egrid, eblk, 0, stream>>>(
            (float4*)y, (const float4*)d_out, (float4*)k1, (const float4*)k7,
            state, N4);
    }

    // final accepted solution
    hipMemcpyAsync(d_out, y, N * sizeof(float), hipMemcpyDeviceToDevice, stream);
}